// BaselinePreprocessor_28741921145370
// MI455X (gfx1250) — compile-verified
//
#include <hip/hip_runtime.h>

// ---------------------------------------------------------------------------
// BaselinePreprocessor for MI455X (gfx1250, wave32).
//  - vision/proprio align-corners interpolation done with V_WMMA_F32_16X16X4_F32:
//    a 16(t) x 16(ch) output tile is W(16x4) @ SRC(4x16) since pos spans <1
//    source row per 16-t tile for vision (K=4) and <5 rows for proprio (two
//    chained K=4 WMMAs = K=8 window).
//  - imu is an exact copy (L == T, align_corners => identity).
//  - voxel grid: zero -> scatter 1.0 -> block reduce -> mean scalar in d_ws.
// Memory-bound: ~144MB stores dominate (~7us at 23.3 TB/s); WMMA keeps the
// interp at 1 matrix instruction per 256 outputs instead of ~512 VALU FMAs.
// ---------------------------------------------------------------------------

typedef __attribute__((ext_vector_type(2))) float v2f;
typedef __attribute__((ext_vector_type(8))) float v8f;

#define VOXG   64
#define NVOX   (VOXG * VOXG * VOXG)
#define NB     64
#define NT     1024
#define CV     512   // vision channels
#define LV     64    // vision source length
#define CP     32    // proprio channels
#define LP     256   // proprio source length
#define CI     6     // imu channels
#define CO     551   // output channels
#define NVTILE (NB * 64 * 32)  // vision wave-tiles: b * t-tile * ch-tile
#define NPTILE (NB * 64 * 2)   // proprio wave-tiles

__device__ __forceinline__ float wcol(int k, int c0, float w) {
    // weight of source-row column k within the K window: (1-w) at c0, w at c0+1
    return (k == c0) ? (1.0f - w) : ((k == c0 + 1) ? w : 0.0f);
}

// --- voxel pipeline ---------------------------------------------------------
__global__ void k_zero_grid(float* __restrict__ ws) {
    int i = blockIdx.x * blockDim.x + threadIdx.x;
    if (i <= NVOX) ws[i] = 0.0f;  // grid + mean slot
}

__global__ void k_scatter(const float* __restrict__ pts, float* __restrict__ grid, int n) {
    int i = blockIdx.x * blockDim.x + threadIdx.x;
    if (i >= n) return;
    float x = pts[3 * i + 0], y = pts[3 * i + 1], z = pts[3 * i + 2];
    int qx = min(max((int)((x + 2.0f) * (VOXG / 4.0f)), 0), VOXG - 1);
    int qy = min(max((int)((y + 2.0f) * (VOXG / 4.0f)), 0), VOXG - 1);
    int qz = min(max((int)((z + 2.0f) * (VOXG / 4.0f)), 0), VOXG - 1);
    grid[(qx * VOXG + qy) * VOXG + qz] = 1.0f;  // all writers store 1.0: race-free
}

__global__ void k_reduce(const float* __restrict__ grid, float* __restrict__ mean_out) {
    __shared__ float s[256];
    float acc = 0.0f;
    for (int i = threadIdx.x; i < NVOX; i += 256) acc += grid[i];
    s[threadIdx.x] = acc;
    __syncthreads();
    for (int off = 128; off > 0; off >>= 1) {
        if (threadIdx.x < off) s[threadIdx.x] += s[threadIdx.x + off];
        __syncthreads();
    }
    if (threadIdx.x == 0) mean_out[0] = s[0] * (1.0f / (float)NVOX);
}

// --- interpolation via WMMA -------------------------------------------------
// One wave32 per 16x16 output tile. Branch between vision/proprio is
// wave-uniform, so EXEC is all-ones at every WMMA (ISA requirement).
__global__ void k_interp_wmma(const float* __restrict__ vision,
                              const float* __restrict__ proprio,
                              float* __restrict__ out) {
    const int lane  = threadIdx.x & 31;
    const int hi16  = lane >> 4;        // 0: lanes 0-15, 1: lanes 16-31
    const int kb    = hi16 * 2;         // K pair held by this lane (A and B frags)
    const int nlo   = lane & 15;        // M for A rows / N for B cols
    const int wid   = blockIdx.x * (blockDim.x >> 5) + (threadIdx.x >> 5);

    if (wid < NVTILE) {
        // ---- vision: one K=4 WMMA per tile ----
        int id = wid;
        int cc = id & 31;  id >>= 5;    // channel tile 0..31
        int tt = id & 63;  id >>= 6;    // t tile 0..63
        int b  = id;
        int t0 = tt * 16;
        int s  = (t0 * (LV - 1)) / (NT - 1);   // exact rational floor

        // A fragment: lane holds A[m][kb], A[m][kb+1]
        int t   = t0 + nlo;
        int num = t * (LV - 1);
        int lo  = num / (NT - 1);
        float w = (float)(num - lo * (NT - 1)) * (1.0f / (float)(NT - 1));
        int c0  = lo - s;                        // 0 or 1 within this tile
        v2f A;
        A.x = wcol(kb,     c0, w);
        A.y = wcol(kb + 1, c0, w);

        // B fragment: lane holds SRC[s+kb][ch], SRC[s+kb+1][ch]
        int ch = cc * 16 + nlo;
        const float* vb = vision + (size_t)b * LV * CV;
        int r0 = min(s + kb,     LV - 1);
        int r1 = min(s + kb + 1, LV - 1);
        v2f Bf;
        Bf.x = vb[(size_t)r0 * CV + ch];
        Bf.y = vb[(size_t)r1 * CV + ch];

        v8f C = {};
        v8f D = __builtin_amdgcn_wmma_f32_16x16x4_f32(false, A, false, Bf,
                                                      (short)0, C, false, false);

        float* ob = out + (size_t)b * NT * CO;
        int trow = t0 + hi16 * 8;
#pragma unroll
        for (int i = 0; i < 8; ++i)
            ob[(size_t)(trow + i) * CO + ch] = D[i];
    } else {
        // ---- proprio: two chained K=4 WMMAs (K window = 8 rows) ----
        int id = wid - NVTILE;
        int cc = id & 1;   id >>= 1;    // channel tile 0..1
        int tt = id & 63;  id >>= 6;    // t tile 0..63
        int b  = id;
        int t0 = tt * 16;
        int s  = (t0 * (LP - 1)) / (NT - 1);

        int t   = t0 + nlo;
        int num = t * (LP - 1);
        int lo  = num / (NT - 1);
        float w = (float)(num - lo * (NT - 1)) * (1.0f / (float)(NT - 1));
        int c0  = lo - s;                        // 0..4 within this tile
        v2f A1, A2;
        A1.x = wcol(kb,     c0, w);  A1.y = wcol(kb + 1, c0, w);
        A2.x = wcol(kb + 4, c0, w);  A2.y = wcol(kb + 5, c0, w);

        int ch = cc * 16 + nlo;
        const float* pb = proprio + (size_t)b * LP * CP;
        int r0 = min(s + kb,     LP - 1);
        int r1 = min(s + kb + 1, LP - 1);
        int r2 = min(s + kb + 4, LP - 1);
        int r3 = min(s + kb + 5, LP - 1);
        v2f B1, B2;
        B1.x = pb[(size_t)r0 * CP + ch];  B1.y = pb[(size_t)r1 * CP + ch];
        B2.x = pb[(size_t)r2 * CP + ch];  B2.y = pb[(size_t)r3 * CP + ch];

        v8f C = {};
        C = __builtin_amdgcn_wmma_f32_16x16x4_f32(false, A1, false, B1,
                                                  (short)0, C, false, false);
        v8f D = __builtin_amdgcn_wmma_f32_16x16x4_f32(false, A2, false, B2,
                                                      (short)0, C, false, false);

        float* ob = out + (size_t)b * NT * CO;
        int trow = t0 + hi16 * 8;
#pragma unroll
        for (int i = 0; i < 8; ++i)
            ob[(size_t)(trow + i) * CO + CV + ch] = D[i];
    }
}

// --- imu copy + summary broadcast ------------------------------------------
__global__ void k_tail(const float* __restrict__ imu, const float* __restrict__ meanp,
                       float* __restrict__ out) {
    int i = blockIdx.x * blockDim.x + threadIdx.x;  // flat (b*NT + t)
    if (i >= NB * NT) return;
    const float* src = imu + (size_t)i * CI;
    float* dst = out + (size_t)i * CO + (CV + CP);
    float m = meanp[0];
#pragma unroll
    for (int k = 0; k < CI; ++k) dst[k] = src[k];
    dst[CI] = m;
}

extern "C" void kernel_launch(void* const* d_in, const int* in_sizes, int n_in,
                              void* d_out, int out_size, void* d_ws, size_t ws_size,
                              hipStream_t stream) {
    const float* vision  = (const float*)d_in[0];
    const float* proprio = (const float*)d_in[1];
    const float* imu     = (const float*)d_in[2];
    // d_in[3] = target_times: unused by the reference computation
    const float* points  = (const float*)d_in[4];
    float* out = (float*)d_out;

    float* grid  = (float*)d_ws;       // NVOX floats
    float* meanp = grid + NVOX;        // 1 float

    int npts = in_sizes[4] / 3;        // 10000

    k_zero_grid<<<(NVOX + 1 + 255) / 256, 256, 0, stream>>>(grid);
    k_scatter<<<(npts + 255) / 256, 256, 0, stream>>>(points, grid, npts);
    k_reduce<<<1, 256, 0, stream>>>(grid, meanp);

    // (NVTILE + NPTILE) wave-tiles, 8 waves per 256-thread block, exact multiple
    k_interp_wmma<<<(NVTILE + NPTILE) / 8, 256, 0, stream>>>(vision, proprio, out);

    k_tail<<<(NB * NT + 255) / 256, 256, 0, stream>>>(imu, meanp, out);
}